// RoIAlignRotated_55310588838018
// MI455X (gfx1250) — compile-verified
//
#include <hip/hip_runtime.h>
#include <hip/hip_bf16.h>

// RoIAlignRotated for MI455X (gfx1250, wave32).
//
// Strategy:
//  - One wave handles one (roi, 16-channel tile). 1000 rois x 16 tiles = 16000 waves.
//  - Lanes 0-15 own channels c+0..15 and compute samples j=0,1 of each bin;
//    lanes 16-31 own the same channels and compute samples j=2,3. That is exactly
//    the documented fp32 16x4 A-matrix VGPR layout for V_WMMA_F32_16X16X4_F32.
//  - The 2x2-sample mean per bin is done with V_WMMA_F32_16X16X4_F32:
//      D[16ch x 16bins] += A[16ch x 4samples] * B[4 x 16bins],
//    B being a one-hot column of 0.25 for the current bin (all 4 K-rows equal,
//    so the fragment is layout-robust). 49 WMMAs per wave.
//  - Features (82 MB) fit in the 192 MB L2, so the ~800 MB of bilinear gather
//    taps are L2 hits; per-lane gathers stay inside the lane's own HxW plane.
//  - Bin loop unrolled 4x: ~32 gathers in flight per wave to cover L2 latency.
//  - Output (50 MB, write-once) stored with non-temporal hint so it does not
//    evict the reused feature planes from L2.

typedef __attribute__((ext_vector_type(2))) float v2f;
typedef __attribute__((ext_vector_type(8))) float v8f;

#define OUT_HW   7
#define FEAT_H   200
#define FEAT_W   200
#define FEAT_C   256
#define SCALE    0.25f

__device__ __forceinline__ float sample_val(const float* __restrict__ fbase,
                                            float yy, float xx,
                                            float sinT, float cosT,
                                            float cy, float cx) {
  // Rotate grid point into feature space.
  float y = yy * cosT - xx * sinT + cy;
  float x = yy * sinT + xx * cosT + cx;

  bool inside = (y >= -1.0f) && (y <= (float)FEAT_H) &&
                (x >= -1.0f) && (x <= (float)FEAT_W);

  float yc = fmaxf(y, 0.0f);
  float xc = fmaxf(x, 0.0f);
  float fy = floorf(yc);
  float fx = floorf(xc);
  int yl = min((int)fy, FEAT_H - 1);
  int xl = min((int)fx, FEAT_W - 1);
  int yh = min(yl + 1, FEAT_H - 1);
  int xh = min(xl + 1, FEAT_W - 1);
  float ly = (fy >= (float)(FEAT_H - 1)) ? 0.0f : (yc - fy);
  float lx = (fx >= (float)(FEAT_W - 1)) ? 0.0f : (xc - fx);
  float hy = 1.0f - ly;
  float hx = 1.0f - lx;
  float w1 = hy * hx, w2 = hy * lx, w3 = ly * hx, w4 = ly * lx;
  if (!inside) { w1 = 0.0f; w2 = 0.0f; w3 = 0.0f; w4 = 0.0f; }

  float v1 = fbase[yl * FEAT_W + xl];
  float v2 = fbase[yl * FEAT_W + xh];
  float v3 = fbase[yh * FEAT_W + xl];
  float v4 = fbase[yh * FEAT_W + xh];
  return w1 * v1 + w2 * v2 + w3 * v3 + w4 * v4;
}

__global__ __launch_bounds__(256)
void RoIAlignRotated_55310588838018_kernel(const float* __restrict__ features,
                                           const float* __restrict__ rois,
                                           float* __restrict__ out,
                                           int nrois) {
  const int roi = blockIdx.x >> 1;
  if (roi >= nrois) return;                 // block-uniform, no divergence
  const int wave   = threadIdx.x >> 5;
  const int lane   = threadIdx.x & 31;
  const int lane16 = lane & 15;
  const int half   = lane >> 4;             // 0: samples j=0,1  1: samples j=2,3
  const int ctile  = ((blockIdx.x & 1) << 3) + wave;   // 0..15

  // ---- per-ROI geometry (wave-uniform values, computed per lane) ----
  const float* r = rois + (size_t)roi * 6;
  const int   b  = (int)r[0];
  const float cx = r[1] * SCALE;
  const float cy = r[2] * SCALE;
  const float rw = fmaxf(r[3] * SCALE, 1.0f);
  const float rh = fmaxf(r[4] * SCALE, 1.0f);
  const float th = r[5];
  const float cosT = cosf(th);
  const float sinT = sinf(th);
  // sample grid: yy = -rh/2 + (sy + 0.5) * rh/14   (out_h*g = 14)
  const float dy = rh * (1.0f / 14.0f);
  const float dx = rw * (1.0f / 14.0f);
  const float y0 = -0.5f * rh;
  const float x0 = -0.5f * rw;

  // ---- per-lane channel plane ----
  const int c = ctile * 16 + lane16;
  const float* fbase = features + ((size_t)(b * FEAT_C + c)) * (FEAT_H * FEAT_W);
  float* outbase = out + ((size_t)roi * FEAT_C + (size_t)ctile * 16) * (OUT_HW * OUT_HW);

  // 49 bins split into 4 WMMA column tiles: [0..15] [16..31] [32..47] [48]
  for (int tile = 0; tile < 4; ++tile) {
    const int nb = (tile < 3) ? 16 : 1;
    v8f acc = {};                           // D/C: 16ch x 16bins fp32
#pragma unroll 4
    for (int bl = 0; bl < nb; ++bl) {
      const int bin = tile * 16 + bl;
      const int py = bin / OUT_HW;
      const int px = bin % OUT_HW;

      // lane computes its two samples of this bin:
      //   sy = py*2 + half ;  sx = px*2 + {0,1}
      const float yy  = y0 + ((float)(py * 2 + half) + 0.5f) * dy;
      const float xxA = x0 + ((float)(px * 2) + 0.5f) * dx;
      const float xxB = xxA + dx;

      v2f a;
      a.x = sample_val(fbase, yy, xxA, sinT, cosT, cy, cx);  // K = 2*half
      a.y = sample_val(fbase, yy, xxB, sinT, cosT, cy, cx);  // K = 2*half+1

      // B[4 x 16]: one-hot column `bl` of 0.25 (all K rows identical).
      const float bw = (lane16 == bl) ? 0.25f : 0.0f;
      v2f bf;
      bf.x = bw;
      bf.y = bw;

      acc = __builtin_amdgcn_wmma_f32_16x16x4_f32(
          /*neg_a=*/false, a, /*neg_b=*/false, bf,
          /*c_mod=*/(short)0, acc, /*reuse_a=*/false, /*reuse_b=*/false);
    }

    // D layout: lane n (n<16) holds bins tile*16+n, rows M = vgpr + 8*half.
    const int bin = tile * 16 + lane16;
    if (bin < OUT_HW * OUT_HW) {
      float* op = outbase + (size_t)(half * 8) * (OUT_HW * OUT_HW) + bin;
#pragma unroll
      for (int rr = 0; rr < 8; ++rr)
        __builtin_nontemporal_store(acc[rr], &op[(size_t)rr * (OUT_HW * OUT_HW)]);
    }
  }
}

extern "C" void kernel_launch(void* const* d_in, const int* in_sizes, int n_in,
                              void* d_out, int out_size, void* d_ws, size_t ws_size,
                              hipStream_t stream) {
  const float* features = (const float*)d_in[0];
  const float* rois     = (const float*)d_in[1];
  float* out            = (float*)d_out;
  const int nrois = in_sizes[1] / 6;        // rois are (N, 6)

  dim3 block(256);                          // 8 waves: 8 channel tiles
  dim3 grid(nrois * 2);                     // 2 blocks per ROI -> 16 tiles
  hipLaunchKernelGGL(RoIAlignRotated_55310588838018_kernel, grid, block, 0, stream,
                     features, rois, out, nrois);
}